// Attention_87497073754486
// MI455X (gfx1250) — compile-verified
//
#include <hip/hip_runtime.h>
#include <math.h>

typedef __attribute__((ext_vector_type(2))) float v2f;
typedef __attribute__((ext_vector_type(8))) float v8f;
typedef __attribute__((ext_vector_type(4))) int   v4i;

#define TB 256
#define MT 128
#define NT 64
#define KC 16

constexpr int Bc    = 16;
constexpr int Hc    = 56;
constexpr int Wc    = 56;
constexpr int Cc    = 256;
constexpr int Gc    = 64;     // C / HEADS
constexpr int C2    = 512;    // C * EXP
constexpr int HWc   = Hc * Wc;        // 3136
constexpr int Ntok  = HWc + 1;        // 3137
constexpr int Mv    = Bc * Ntok;      // 50192 rows (with CLS)
constexpr int Ms    = Bc * HWc;       // 50176 rows (spatial only)
constexpr int BNROWS = 64;
constexpr int NBLK   = Ms / BNROWS;   // 784
constexpr float EPSc = 1e-5f;

// ---- gfx1250 async global->LDS path (guarded; falls back to direct stores) --
#if defined(__HIP_DEVICE_COMPILE__) && __has_builtin(__builtin_amdgcn_global_load_async_to_lds_b128)
#define HAVE_ASYNC_LDS 1
using gv4i = __attribute__((address_space(1))) v4i;   // global v4i
using lv4i = __attribute__((address_space(3))) v4i;   // LDS v4i
#else
#define HAVE_ASYNC_LDS 0
#endif

template <int N>
__device__ __forceinline__ void wait_asynccnt_imm() {
#if defined(__HIP_DEVICE_COMPILE__)
#if __has_builtin(__builtin_amdgcn_s_wait_asynccnt)
  __builtin_amdgcn_s_wait_asynccnt(N);
#else
  asm volatile("s_wait_asynccnt %0" ::"n"(N));
#endif
#endif
}

__device__ __forceinline__ float gelu_exact(float t) {
  return 0.5f * t * (1.0f + erff(t * 0.7071067811865475f));
}

// -----------------------------------------------------------------------------
// Generic f32 WMMA GEMM:  OUT[m, n] = sum_k A[m,k] * Wm[n,k]  (+ bias[n])
//   AROWMODE: 0 = linear rows; 1 = skip CLS token per batch (row = b*Nfull+1+hw)
//   OUTMODE : 0 = row-major (M, Nout); 1 = channel-major (b, n, hw)
//   BNGELU  : transform A elements with per-channel BN + exact GELU (sync path)
// Tiles: 128(M) x 64(N), 8 waves; wave owns a 16x64 strip (4 accumulators).
// Non-BNGELU path: double-buffered GLOBAL_LOAD_ASYNC_TO_LDS_B128 pipeline.
// -----------------------------------------------------------------------------
template <int AROWMODE, int OUTMODE, int BNGELU, int KTOT>
__global__ __launch_bounds__(TB) void gemm_f32_wmma(
    const float* __restrict__ A, const float* __restrict__ Wm,
    const float* __restrict__ bias, float* __restrict__ OUT,
    int M, int Nout, int HW, int NfullRows,
    const float* __restrict__ bnMean, const float* __restrict__ bnRstd,
    const float* __restrict__ bnG, const float* __restrict__ bnB)
{
  constexpr int K   = KTOT;
  constexpr int NCH = KTOT / KC;

  __shared__ __align__(16) float As[2][MT][KC + 4];
  __shared__ __align__(16) float Bs[2][NT][KC + 4];

  const int tid  = threadIdx.x;
  const int lane = tid & 31;
  const int wave = tid >> 5;
  const int lo   = lane & 15;
  const int hi   = lane >> 4;
  const int m0   = blockIdx.x * MT;
  const int n0   = blockIdx.y * NT;

  // Per-thread tile slots (fixed across K chunks).
  const int rowA0 = tid >> 2;          // slot tid       -> A rows 0..63
  const int rowA1 = (tid + TB) >> 2;   // slot tid+256   -> A rows 64..127
  const int q     = tid & 3;           // float4 quad within the 16-wide row
  const int rowB  = tid >> 2;

  // Out-of-range rows are clamped (their outputs are never stored; an A row
  // only ever contributes to its own output row, so this is safe).
  int mA0 = m0 + rowA0; if (mA0 >= M) mA0 = M - 1;
  int mA1 = m0 + rowA1; if (mA1 >= M) mA1 = M - 1;

  long long ar0, ar1;
  if constexpr (AROWMODE == 1) {
    int b0 = mA0 / HW; ar0 = (long long)b0 * NfullRows + 1 + (mA0 - b0 * HW);
    int b1 = mA1 / HW; ar1 = (long long)b1 * NfullRows + 1 + (mA1 - b1 * HW);
  } else {
    ar0 = mA0; ar1 = mA1;
  }
  const float* aSrc0 = A + ar0 * (long long)K + q * 4;
  const float* aSrc1 = A + ar1 * (long long)K + q * 4;
  const float* bSrc  = Wm + (long long)(n0 + rowB) * K + q * 4;

  v8f acc[4] = {};

  auto compute = [&](int buf) {
    // A frag (16x4 f32): lane lo = M row, VGPR j holds K = 2*hi + j
    // B frag (4x16 f32): lane lo = N col, VGPR j holds K = 2*hi + j
#pragma unroll
    for (int kk = 0; kk < KC; kk += 4) {
      v2f a;
      a.x = As[buf][wave * 16 + lo][kk + 2 * hi + 0];
      a.y = As[buf][wave * 16 + lo][kk + 2 * hi + 1];
#pragma unroll
      for (int nt = 0; nt < 4; ++nt) {
        v2f b;
        b.x = Bs[buf][nt * 16 + lo][kk + 2 * hi + 0];
        b.y = Bs[buf][nt * 16 + lo][kk + 2 * hi + 1];
        acc[nt] = __builtin_amdgcn_wmma_f32_16x16x4_f32(
            false, a, false, b, (short)0, acc[nt], false, false);
      }
    }
  };

  if constexpr (!BNGELU) {
    auto issue = [&](int k0, int buf) {
#if HAVE_ASYNC_LDS
      __builtin_amdgcn_global_load_async_to_lds_b128(
          (gv4i*)(aSrc0 + k0), (lv4i*)&As[buf][rowA0][q * 4], 0, 0);
      __builtin_amdgcn_global_load_async_to_lds_b128(
          (gv4i*)(aSrc1 + k0), (lv4i*)&As[buf][rowA1][q * 4], 0, 0);
      __builtin_amdgcn_global_load_async_to_lds_b128(
          (gv4i*)(bSrc + k0), (lv4i*)&Bs[buf][rowB][q * 4], 0, 0);
#else
      *(float4*)&As[buf][rowA0][q * 4] = *(const float4*)(aSrc0 + k0);
      *(float4*)&As[buf][rowA1][q * 4] = *(const float4*)(aSrc1 + k0);
      *(float4*)&Bs[buf][rowB][q * 4]  = *(const float4*)(bSrc + k0);
#endif
    };

    issue(0, 0);
#pragma unroll 1
    for (int i = 0; i < NCH; ++i) {
      const int buf = i & 1;
      if (i + 1 < NCH) {
        issue((i + 1) * KC, buf ^ 1);  // prefetch next chunk into other buffer
        wait_asynccnt_imm<3>();        // retire this chunk's 3 async b128s
      } else {
        wait_asynccnt_imm<0>();
      }
      __syncthreads();
      compute(buf);
      __syncthreads();
    }
  } else {
    // Synchronous staging: BN + exact GELU applied in VALU before LDS store.
#pragma unroll 1
    for (int k0 = 0; k0 < K; k0 += KC) {
      float4 v0 = *(const float4*)(aSrc0 + k0);
      float4 v1 = *(const float4*)(aSrc1 + k0);
      float4 wb = *(const float4*)(bSrc + k0);
      const int c  = k0 + q * 4;
      float4 gm = *(const float4*)(bnG + c);
      float4 mu = *(const float4*)(bnMean + c);
      float4 rs = *(const float4*)(bnRstd + c);
      float4 bt = *(const float4*)(bnB + c);
      v0.x = gelu_exact(gm.x * (v0.x - mu.x) * rs.x + bt.x);
      v0.y = gelu_exact(gm.y * (v0.y - mu.y) * rs.y + bt.y);
      v0.z = gelu_exact(gm.z * (v0.z - mu.z) * rs.z + bt.z);
      v0.w = gelu_exact(gm.w * (v0.w - mu.w) * rs.w + bt.w);
      v1.x = gelu_exact(gm.x * (v1.x - mu.x) * rs.x + bt.x);
      v1.y = gelu_exact(gm.y * (v1.y - mu.y) * rs.y + bt.y);
      v1.z = gelu_exact(gm.z * (v1.z - mu.z) * rs.z + bt.z);
      v1.w = gelu_exact(gm.w * (v1.w - mu.w) * rs.w + bt.w);
      *(float4*)&As[0][rowA0][q * 4] = v0;
      *(float4*)&As[0][rowA1][q * 4] = v1;
      *(float4*)&Bs[0][rowB][q * 4]  = wb;
      __syncthreads();
      compute(0);
      __syncthreads();
    }
  }

  // ---- store: C/D layout: lane lo = N, VGPR r -> M = r + 8*hi ----
#pragma unroll
  for (int nt = 0; nt < 4; ++nt) {
    int n = n0 + nt * 16 + lo;
    float bv = bias ? bias[n] : 0.0f;
#pragma unroll
    for (int r = 0; r < 8; ++r) {
      int m = m0 + wave * 16 + r + 8 * hi;
      if (m < M) {
        float val = acc[nt][r] + bv;
        if constexpr (OUTMODE == 0) {
          OUT[(long long)m * Nout + n] = val;
        } else {
          int b = m / HW; int hw = m - b * HW;
          OUT[((long long)b * Nout + n) * HW + hw] = val;
        }
      }
    }
  }
}

// -----------------------------------------------------------------------------
// Fused per-head depthwise conv (k = 3,5,7,9) + grouped 1x1 (4 heads -> 8 out)
// S layout: (B, C, HW) channel-major (c = head*64 + g)
// Y layout: (B*HW, 512) row-major, oc = g*8 + j
// -----------------------------------------------------------------------------
__global__ __launch_bounds__(TB) void dwconv_p0_kernel(
    const float* __restrict__ S,
    const float* __restrict__ lw0, const float* __restrict__ lb0,
    const float* __restrict__ lw1, const float* __restrict__ lb1,
    const float* __restrict__ lw2, const float* __restrict__ lb2,
    const float* __restrict__ lw3, const float* __restrict__ lb3,
    const float* __restrict__ p0w, const float* __restrict__ p0b,
    float* __restrict__ Y)
{
  int hw = blockIdx.x * TB + threadIdx.x;
  int bg = blockIdx.y;
  int b  = bg >> 6;
  int g  = bg & 63;
  if (hw >= HWc) return;
  int h = hw / Wc, w = hw - h * Wc;

  const float* lws[4] = {lw0, lw1, lw2, lw3};
  const float* lbs[4] = {lb0, lb1, lb2, lb3};
  float d[4];
#pragma unroll
  for (int head = 0; head < 4; ++head) {
    const int ks = 3 + 2 * head, pad = 1 + head;
    const float* base = S + (long long)(b * Cc + head * Gc + g) * HWc;
    const float* wp   = lws[head] + g * ks * ks;
    float acc = lbs[head][g];
    for (int ky = 0; ky < ks; ++ky) {
      int yy = h + ky - pad;
      if (yy < 0 || yy >= Hc) continue;
      for (int kx = 0; kx < ks; ++kx) {
        int xx = w + kx - pad;
        if (xx < 0 || xx >= Wc) continue;
        acc += base[yy * Wc + xx] * wp[ky * ks + kx];
      }
    }
    d[head] = acc;
  }

  float* yrow = Y + (long long)(b * HWc + hw) * C2 + g * 8;
#pragma unroll
  for (int j = 0; j < 8; ++j) {
    int oc = g * 8 + j;
    float val = p0b[oc];
#pragma unroll
    for (int i = 0; i < 4; ++i) val += p0w[oc * 4 + i] * d[i];
    yrow[j] = val;
  }
}

// -----------------------------------------------------------------------------
// Deterministic BatchNorm statistics (no float atomics -> replay-stable).
// -----------------------------------------------------------------------------
__global__ __launch_bounds__(TB) void bn_partial_kernel(
    const float* __restrict__ Y, float* __restrict__ P)
{
  int r0 = blockIdx.x * BNROWS;
  int c0 = threadIdx.x;                 // handles channels c0 and c0+256
  float s0 = 0.f, q0 = 0.f, s1 = 0.f, q1 = 0.f;
  for (int r = 0; r < BNROWS; ++r) {
    const float* row = Y + (long long)(r0 + r) * C2;
    float a = row[c0];
    float b = row[c0 + 256];
    s0 += a; q0 += a * a;
    s1 += b; q1 += b * b;
  }
  float* p = P + (long long)blockIdx.x * (2 * C2);
  p[c0]            = s0;
  p[c0 + 256]      = s1;
  p[C2 + c0]       = q0;
  p[C2 + c0 + 256] = q1;
}

__global__ __launch_bounds__(TB) void bn_finalize_kernel(
    const float* __restrict__ P, float* __restrict__ meanOut,
    float* __restrict__ rstdOut)
{
  int c = blockIdx.x * TB + threadIdx.x;
  if (c >= C2) return;
  float s = 0.f, q = 0.f;
  for (int b = 0; b < NBLK; ++b) {
    s += P[(long long)b * (2 * C2) + c];
    q += P[(long long)b * (2 * C2) + C2 + c];
  }
  const float cnt = (float)Ms;
  float mean = s / cnt;
  float var  = q / cnt - mean * mean;
  meanOut[c] = mean;
  rstdOut[c] = rsqrtf(var + EPSc);
}

// -----------------------------------------------------------------------------
// u[b,n,c] = (n==0 ? x[b,0,c] : z[b,n-1,c]) * v[b,n,c]
// -----------------------------------------------------------------------------
__global__ __launch_bounds__(TB) void build_u_kernel(
    const float* __restrict__ x, const float* __restrict__ v,
    const float* __restrict__ z, float* __restrict__ U)
{
  const long long total = (long long)Mv * Cc;
  for (long long e = (long long)blockIdx.x * TB + threadIdx.x; e < total;
       e += (long long)gridDim.x * TB) {
    long long b = e / ((long long)Ntok * Cc);
    int rem = (int)(e - b * ((long long)Ntok * Cc));
    int n = rem / Cc;
    int c = rem - n * Cc;
    float base = (n == 0) ? x[e]
                          : z[((long long)b * HWc + (n - 1)) * Cc + c];
    U[e] = base * v[e];
  }
}

// -----------------------------------------------------------------------------
extern "C" void kernel_launch(void* const* d_in, const int* in_sizes, int n_in,
                              void* d_out, int out_size, void* d_ws, size_t ws_size,
                              hipStream_t stream) {
  (void)in_sizes; (void)n_in; (void)out_size; (void)ws_size;

  const float* x   = (const float*)d_in[0];
  // d_in[1] = H, d_in[2] = W (compile-time constants here)
  const float* Wv  = (const float*)d_in[3];
  const float* Ws_ = (const float*)d_in[4];
  const float* lw0 = (const float*)d_in[5];
  const float* lb0 = (const float*)d_in[6];
  const float* lw1 = (const float*)d_in[7];
  const float* lb1 = (const float*)d_in[8];
  const float* lw2 = (const float*)d_in[9];
  const float* lb2 = (const float*)d_in[10];
  const float* lw3 = (const float*)d_in[11];
  const float* lb3 = (const float*)d_in[12];
  const float* p0w = (const float*)d_in[13];
  const float* p0b = (const float*)d_in[14];
  const float* bng = (const float*)d_in[15];
  const float* bnb = (const float*)d_in[16];
  const float* p1w = (const float*)d_in[17];
  const float* p1b = (const float*)d_in[18];
  const float* Wp  = (const float*)d_in[19];
  const float* bp  = (const float*)d_in[20];

  float* out = (float*)d_out;
  float* ws  = (float*)d_ws;

  size_t offV    = 0;
  size_t offS    = offV + (size_t)Mv * Cc;          // v  : 50192*256
  size_t offY    = offS + (size_t)Ms * Cc;          // s  : 50176*256 (B,C,HW)
  size_t offZ    = offY + (size_t)Ms * C2;          // y  : 50176*512
  size_t offP    = offZ + (size_t)Ms * Cc;          // z  : 50176*256
  size_t offMean = offP + (size_t)NBLK * 2 * C2;    // partials
  size_t offRstd = offMean + C2;

  float* v    = ws + offV;
  float* s    = ws + offS;
  float* y    = ws + offY;
  float* z    = ws + offZ;
  float* part = ws + offP;
  float* mean = ws + offMean;
  float* rstd = ws + offRstd;
  float* u    = s;  // s is dead after the conv kernel

  dim3 blk(TB);

  // 1) v = x @ Wv^T  (row-major out)
  gemm_f32_wmma<0, 0, 0, Cc><<<dim3((Mv + MT - 1) / MT, Cc / NT), blk, 0, stream>>>(
      x, Wv, nullptr, v, Mv, Cc, HWc, Ntok, nullptr, nullptr, nullptr, nullptr);

  // 2) s = x[:,1:,:] @ Ws^T  (channel-major out for the convs)
  gemm_f32_wmma<1, 1, 0, Cc><<<dim3(Ms / MT, Cc / NT), blk, 0, stream>>>(
      x, Ws_, nullptr, s, Ms, Cc, HWc, Ntok, nullptr, nullptr, nullptr, nullptr);

  // 3) multi-scale depthwise conv + grouped p0 mix -> y (B*HW, 512)
  dwconv_p0_kernel<<<dim3((HWc + TB - 1) / TB, Bc * Gc), blk, 0, stream>>>(
      s, lw0, lb0, lw1, lb1, lw2, lb2, lw3, lb3, p0w, p0b, y);

  // 4) deterministic BN stats
  bn_partial_kernel<<<dim3(NBLK), blk, 0, stream>>>(y, part);
  bn_finalize_kernel<<<dim3((C2 + TB - 1) / TB), blk, 0, stream>>>(part, mean, rstd);

  // 5) z = GELU(BN(y)) @ p1^T + p1b   (BN+GELU fused into A-tile load)
  gemm_f32_wmma<0, 0, 1, C2><<<dim3(Ms / MT, Cc / NT), blk, 0, stream>>>(
      y, p1w, p1b, z, Ms, Cc, HWc, Ntok, mean, rstd, bng, bnb);

  // 6) u = concat(tok, z) * v
  build_u_kernel<<<dim3(2048), blk, 0, stream>>>(x, v, z, u);

  // 7) out = u @ Wp^T + bp
  gemm_f32_wmma<0, 0, 0, Cc><<<dim3((Mv + MT - 1) / MT, Cc / NT), blk, 0, stream>>>(
      u, Wp, bp, out, Mv, Cc, HWc, Ntok, nullptr, nullptr, nullptr, nullptr);
}